// MSDAv2_46394236731950
// MI455X (gfx1250) — compile-verified
//
#include <hip/hip_runtime.h>
#include <hip/hip_bf16.h>
#include <math.h>

// ---------------------------------------------------------------------------
// MSDAv2 on gfx1250 (MI455X): fp32 end-to-end.
//   GEMMs via V_WMMA_F32_16X16X4_F32, one wave32 per 32x32 output tile
//   (2 A-frags x 2 B-frags -> 4 WMMAs per k-step; 2x operand reuse vs 16x16).
//   Bilinear deformable sampling: one wave per (b,q,head); lane == channel.
// Constants from the reference: B=2, D=256, H=8, L=5, P=4, Dh=32.
// ---------------------------------------------------------------------------

typedef float v2f __attribute__((ext_vector_type(2)));
typedef float v8f __attribute__((ext_vector_type(8)));

#define MSDA_B   2
#define MSDA_D   256
#define MSDA_H   8
#define MSDA_L   5
#define MSDA_P   4
#define MSDA_DH  32
#define MSDA_LP  (MSDA_L * MSDA_P)   // 20

// ---------------------------------------------------------------------------
// C = A * W^T + bias  with f32 WMMA (16x16x4), 32x32 tile per wave.
//   A:    [M, K] row-major (K multiple of 4)
//   W:    [N, K] row-major (so B-matrix B[k][n] = W[n][k])
//   bias: [N]
//   C:    [M, N] row-major  (N must be a multiple of 32)
// A-frag layout (16x4 f32): both half-waves hold M=lane&15;
//   VGPR0 = K {0 | 2}, VGPR1 = K {1 | 3} for halves {lo | hi}.
// B-frag (4x16) mirrors it with N=lane&15.
// D layout: VGPR i -> M = i + 8*(lane>=16), N = lane&15.
// ---------------------------------------------------------------------------
__global__ __launch_bounds__(32)
void msda_wmma_gemm_f32(const float* __restrict__ A,
                        const float* __restrict__ W,
                        const float* __restrict__ bias,
                        float* __restrict__ C,
                        int M, int N, int K)
{
    const int ntn  = N >> 5;                  // tiles along N (32 wide)
    const int tile = blockIdx.x;
    const int tm   = tile / ntn;
    const int tn   = tile - tm * ntn;

    const int lane = threadIdx.x & 31;
    const int half = lane >> 4;               // 0: K+{0,1}, 1: K+{2,3}
    const int l16  = lane & 15;

    int ar0 = tm * 32 + l16;
    int ar1 = tm * 32 + 16 + l16;
    if (ar0 >= M) ar0 = M - 1;                // clamp loads; stores guarded below
    if (ar1 >= M) ar1 = M - 1;
    const int br0 = tn * 32 + l16;            // always < N (N multiple of 32)
    const int br1 = tn * 32 + 16 + l16;

    const float* __restrict__ ap0 = A + (size_t)ar0 * K + 2 * half;
    const float* __restrict__ ap1 = A + (size_t)ar1 * K + 2 * half;
    const float* __restrict__ bp0 = W + (size_t)br0 * K + 2 * half;
    const float* __restrict__ bp1 = W + (size_t)br1 * K + 2 * half;

    v8f acc00 = {}, acc01 = {}, acc10 = {}, acc11 = {};

#pragma unroll 4
    for (int k0 = 0; k0 < K; k0 += 4) {
        v2f a0, a1, b0, b1;
        a0.x = ap0[k0]; a0.y = ap0[k0 + 1];
        a1.x = ap1[k0]; a1.y = ap1[k0 + 1];
        b0.x = bp0[k0]; b0.y = bp0[k0 + 1];
        b1.x = bp1[k0]; b1.y = bp1[k0 + 1];
        acc00 = __builtin_amdgcn_wmma_f32_16x16x4_f32(false, a0, false, b0,
                                                      (short)0, acc00, false, false);
        acc01 = __builtin_amdgcn_wmma_f32_16x16x4_f32(false, a0, false, b1,
                                                      (short)0, acc01, false, false);
        acc10 = __builtin_amdgcn_wmma_f32_16x16x4_f32(false, a1, false, b0,
                                                      (short)0, acc10, false, false);
        acc11 = __builtin_amdgcn_wmma_f32_16x16x4_f32(false, a1, false, b1,
                                                      (short)0, acc11, false, false);
    }

    const int n0 = tn * 32 + l16;
    const int n1 = tn * 32 + 16 + l16;
    const float bv0 = bias[n0];
    const float bv1 = bias[n1];
    const int mbase = tm * 32 + half * 8;

#pragma unroll
    for (int i = 0; i < 8; ++i) {
        const int m0 = mbase + i;             // rows of the upper 16-row band
        const int m1 = mbase + 16 + i;        // rows of the lower 16-row band
        if (m0 < M) {
            C[(size_t)m0 * N + n0] = acc00[i] + bv0;
            C[(size_t)m0 * N + n1] = acc01[i] + bv1;
        }
        if (m1 < M) {
            C[(size_t)m1 * N + n0] = acc10[i] + bv0;
            C[(size_t)m1 * N + n1] = acc11[i] + bv1;
        }
    }
}

// ---------------------------------------------------------------------------
// Deformable bilinear sampling + attention-weighted head reduction.
// Block = one (b, q): 256 threads = 8 heads x 32 channels.
//   off_buf : [B*Nq, H*L*P*2]  (h-major, then l, then p, then xy)
//   attn_buf: [B*Nq, H*L*P]    (pre-softmax logits)
//   priors  : [B, Nq, L, 2]
//   val     : [B, S, 256]      (head h owns channels [h*32, h*32+32))
//   tmp     : [B, Nq, 256]     output (input of the final GEMM)
// ---------------------------------------------------------------------------
__global__ __launch_bounds__(256)
void msda_sample_kernel(const float* __restrict__ off_buf,
                        const float* __restrict__ attn_buf,
                        const float* __restrict__ priors,
                        const float* __restrict__ val,
                        const int*   __restrict__ shapes_hw,  // [L][2] = (H_l, W_l)
                        const int*   __restrict__ start_ids,  // [L]
                        float* __restrict__ tmp,
                        int Nq, int S)
{
    const int bq = blockIdx.x;              // 0 .. B*Nq-1
    const int b  = bq / Nq;
    const int h  = threadIdx.x >> 5;        // head
    const int c  = threadIdx.x & 31;        // channel within head

    // ---- softmax over the L*P = 20 logits of this head (broadcast loads) ----
    const float* __restrict__ logit = attn_buf + (size_t)bq * (MSDA_H * MSDA_LP) + h * MSDA_LP;
    float e[MSDA_LP];
    float mx = logit[0];
#pragma unroll
    for (int j = 1; j < MSDA_LP; ++j) mx = fmaxf(mx, logit[j]);
    float sum = 0.0f;
#pragma unroll
    for (int j = 0; j < MSDA_LP; ++j) { e[j] = expf(logit[j] - mx); sum += e[j]; }
    const float inv = 1.0f / sum;

    const float* __restrict__ offr  = off_buf + (size_t)bq * (MSDA_H * MSDA_LP * 2) + h * (MSDA_LP * 2);
    const float* __restrict__ pr    = priors  + (size_t)bq * (MSDA_L * 2);
    const float* __restrict__ vbase = val + (size_t)b * S * MSDA_D + h * MSDA_DH + c;

    float acc = 0.0f;
#pragma unroll
    for (int l = 0; l < MSDA_L; ++l) {
        const int   Hi = shapes_hw[l * 2 + 0];
        const int   Wi = shapes_hw[l * 2 + 1];
        const int   st = start_ids[l];
        const float fw = (float)Wi, fh = (float)Hi;
        const float px = pr[l * 2 + 0], py = pr[l * 2 + 1];
#pragma unroll
        for (int p = 0; p < MSDA_P; ++p) {
            const float aw = e[l * MSDA_P + p] * inv;
            const float ox = offr[(l * MSDA_P + p) * 2 + 0];
            const float oy = offr[(l * MSDA_P + p) * 2 + 1];
            // locs = prior + off / (W,H); then x = loc_x*W - 0.5 (per reference)
            const float x = (px + ox / fw) * fw - 0.5f;
            const float y = (py + oy / fh) * fh - 0.5f;
            const float x0f = floorf(x), y0f = floorf(y);
            const int   x0  = (int)x0f,  y0  = (int)y0f;
            const float wx1 = x - x0f,   wy1 = y - y0f;
            const float wx0 = 1.0f - wx1, wy0 = 1.0f - wy1;
#pragma unroll
            for (int dy = 0; dy < 2; ++dy) {
#pragma unroll
                for (int dx = 0; dx < 2; ++dx) {
                    const int xi = x0 + dx, yi = y0 + dy;
                    const float w = (dx ? wx1 : wx0) * (dy ? wy1 : wy0);
                    const bool valid = (xi >= 0) & (xi < Wi) & (yi >= 0) & (yi < Hi);
                    const int xc = min(max(xi, 0), Wi - 1);
                    const int yc = min(max(yi, 0), Hi - 1);
                    const int idx = st + yc * Wi + xc;
                    const float v = vbase[(size_t)idx * MSDA_D];   // coalesced: lane==channel
                    acc += valid ? (aw * w * v) : 0.0f;
                }
            }
        }
    }
    tmp[(size_t)bq * MSDA_D + h * MSDA_DH + c] = acc;
}

// ---------------------------------------------------------------------------
extern "C" void kernel_launch(void* const* d_in, const int* in_sizes, int n_in,
                              void* d_out, int out_size, void* d_ws, size_t ws_size,
                              hipStream_t stream)
{
    const float* in_feats   = (const float*)d_in[0];   // [B, Nq, 256]
    const float* priors     = (const float*)d_in[1];   // [B, Nq, L, 2]
    const float* samp_feats = (const float*)d_in[2];   // [B, S, 256]
    const float* W_off      = (const float*)d_in[3];   // [320, 256]
    const float* b_off      = (const float*)d_in[4];   // [320]
    const float* W_attn     = (const float*)d_in[5];   // [160, 256]
    const float* b_attn     = (const float*)d_in[6];   // [160]
    const float* W_val      = (const float*)d_in[7];   // [256, 256]
    const float* b_val      = (const float*)d_in[8];   // [256]
    const float* W_out      = (const float*)d_in[9];   // [256, 256]
    const float* b_out      = (const float*)d_in[10];  // [256]
    const int*   shapes_hw  = (const int*)d_in[11];    // [L, 2]
    const int*   start_ids  = (const int*)d_in[12];    // [L]
    float*       out        = (float*)d_out;

    const int Nq = in_sizes[0] / (MSDA_B * MSDA_D);
    const int S  = in_sizes[2] / (MSDA_B * MSDA_D);
    const int BN = MSDA_B * Nq;                         // GEMM M for query-side
    const int BS = MSDA_B * S;                          // GEMM M for value-side

    const int NOFF  = MSDA_H * MSDA_LP * 2;             // 320
    const int NATTN = MSDA_H * MSDA_LP;                 // 160

    // workspace carve (floats): val | off_buf | attn_buf | tmp
    float* ws       = (float*)d_ws;
    float* val      = ws;
    float* off_buf  = val      + (size_t)BS * MSDA_D;
    float* attn_buf = off_buf  + (size_t)BN * NOFF;
    float* tmp      = attn_buf + (size_t)BN * NATTN;

    const int tmM_bs = (BS + 31) / 32;                  // 32-row tiles
    const int tmM_bn = (BN + 31) / 32;

    // 1) val = sample_feats @ W_val.T + b_val
    msda_wmma_gemm_f32<<<tmM_bs * (MSDA_D / 32), 32, 0, stream>>>(
        samp_feats, W_val, b_val, val, BS, MSDA_D, MSDA_D);

    // 2) offsets = in_feats @ W_off.T + b_off
    msda_wmma_gemm_f32<<<tmM_bn * (NOFF / 32), 32, 0, stream>>>(
        in_feats, W_off, b_off, off_buf, BN, NOFF, MSDA_D);

    // 3) attn logits = in_feats @ W_attn.T + b_attn
    msda_wmma_gemm_f32<<<tmM_bn * (NATTN / 32), 32, 0, stream>>>(
        in_feats, W_attn, b_attn, attn_buf, BN, NATTN, MSDA_D);

    // 4) softmax + deformable bilinear sampling + head-weighted reduction
    msda_sample_kernel<<<BN, 256, 0, stream>>>(
        off_buf, attn_buf, priors, val, shapes_hw, start_ids, tmp, Nq, S);

    // 5) out = tmp @ W_out.T + b_out
    msda_wmma_gemm_f32<<<tmM_bn * (MSDA_D / 32), 32, 0, stream>>>(
        tmp, W_out, b_out, out, BN, MSDA_D, MSDA_D);
}